// LocalDynamics_71871982731546
// MI455X (gfx1250) — compile-verified
//
#include <hip/hip_runtime.h>

#define Bn 8
#define Cc 64
#define HW 4096          // 64*64
#define NT (HW / 16)     // 256 support tiles

typedef __attribute__((ext_vector_type(2))) float v2f;
typedef __attribute__((ext_vector_type(8))) float v8f;

// ---------------------------------------------------------------------------
// Kernel 1: fused simi GEMM (q^T s, K=64) + row-wise max/argmax.
// 8 waves per block, one 16-query-row tile per wave; the block shares a
// double-buffered LDS s-tile so global->LDS staging overlaps the WMMA chain.
// B tile preloaded to registers (batched ds_loads, single wait), and the
// K-accumulation split across two independent WMMA accumulator chains.
// ---------------------------------------------------------------------------
__global__ __launch_bounds__(256) void simi_argmax_kernel(
    const float* __restrict__ x, const float* __restrict__ mask,
    float* __restrict__ wv, int* __restrict__ idxout)
{
    __shared__ float sT[2][Cc][16];              // double-buffered s tile, 8 KB

    const int b    = blockIdx.y;
    const int wave = threadIdx.x >> 5;
    const int lane = threadIdx.x & 31;
    const int half = lane >> 4;                  // 0: lanes 0-15, 1: lanes 16-31
    const int l16  = lane & 15;
    const int n0   = (blockIdx.x * 8 + wave) * 16;

    const float* xb = x    + (size_t)b * Cc * HW;
    const float* mb = mask + (size_t)b * HW;

    // Preload A strip: q rows n0..n0+15, K=0..63.
    // 16x4 fp32 A layout: lane m holds K = 2*half + {0,1} per k-step.
    v2f A[16];
    {
        const float mq = mb[n0 + l16];
        #pragma unroll
        for (int t = 0; t < 16; ++t) {
            const int k = 4 * t + 2 * half;
            A[t].x = xb[(size_t)k       * HW + n0 + l16] * mq;
            A[t].y = xb[(size_t)(k + 1) * HW + n0 + l16] * mq;
        }
    }

    float vmax[8];
    int   vidx[8];
    #pragma unroll
    for (int i = 0; i < 8; ++i) { vmax[i] = -3.402823466e38f; vidx[i] = 0; }

    // Prologue: stage tile 0 into buffer 0 (4 elements per thread).
    float sreg[4];
    #pragma unroll
    for (int i = 0; i < 4; ++i) {
        const int e = threadIdx.x + i * 256;
        const int c = e >> 4, j = e & 15;
        sreg[i] = xb[(size_t)c * HW + j] * (1.0f - mb[j]);
    }
    #pragma unroll
    for (int i = 0; i < 4; ++i) {
        const int e = threadIdx.x + i * 256;
        sT[0][e >> 4][e & 15] = sreg[i];
    }
    __syncthreads();

    for (int mt = 0; mt < NT; ++mt) {
        const int buf = mt & 1;
        const int m0  = mt * 16;
        const bool more = (mt + 1) < NT;

        // Issue next tile's global loads early; latency hides behind WMMAs.
        if (more) {
            const int mn0 = m0 + 16;
            #pragma unroll
            for (int i = 0; i < 4; ++i) {
                const int e = threadIdx.x + i * 256;
                const int c = e >> 4, j = e & 15;
                sreg[i] = xb[(size_t)c * HW + mn0 + j] * (1.0f - mb[mn0 + j]);
            }
        }

        // Preload entire B tile into registers: all ds_loads issue together,
        // so the WMMA chain below runs without interleaved LDS waits.
        v2f Bv[16];
        #pragma unroll
        for (int t = 0; t < 16; ++t) {
            const int k = 4 * t + 2 * half;
            Bv[t].x = sT[buf][k][l16];
            Bv[t].y = sT[buf][k + 1][l16];
        }

        // simi tile = A(16x64) * B(64x16): two independent 8-deep WMMA chains.
        v8f acc0 = {};
        v8f acc1 = {};
        #pragma unroll
        for (int t = 0; t < 8; ++t) {
            acc0 = __builtin_amdgcn_wmma_f32_16x16x4_f32(
                false, A[2 * t],     false, Bv[2 * t],     (short)0, acc0, false, false);
            acc1 = __builtin_amdgcn_wmma_f32_16x16x4_f32(
                false, A[2 * t + 1], false, Bv[2 * t + 1], (short)0, acc1, false, false);
        }

        // Running max/argmax. D layout: VGPR i = row i (half 0) / i+8 (half 1),
        // lane%16 = column. Strict '>' keeps first occurrence within a lane.
        const int col = m0 + l16;
        #pragma unroll
        for (int i = 0; i < 8; ++i) {
            const float v = acc0[i] + acc1[i];
            if (v > vmax[i]) { vmax[i] = v; vidx[i] = col; }
        }

        // Store staged tile into the alternate buffer, single barrier/iter.
        if (more) {
            #pragma unroll
            for (int i = 0; i < 4; ++i) {
                const int e = threadIdx.x + i * 256;
                sT[buf ^ 1][e >> 4][e & 15] = sreg[i];
            }
        }
        __syncthreads();
    }

    // Cross-lane (max, idx) reduce within each 16-lane half; ties -> lower idx.
    #pragma unroll
    for (int off = 8; off >= 1; off >>= 1) {
        #pragma unroll
        for (int i = 0; i < 8; ++i) {
            const float om = __shfl_xor(vmax[i], off, 32);
            const int   oi = __shfl_xor(vidx[i], off, 32);
            if (om > vmax[i] || (om == vmax[i] && oi < vidx[i])) {
                vmax[i] = om; vidx[i] = oi;
            }
        }
    }

    if (l16 == 0) {
        #pragma unroll
        for (int i = 0; i < 8; ++i) {
            const int n = n0 + i + 8 * half;
            wv[(size_t)b * HW + n]     = vmax[i];
            idxout[(size_t)b * HW + n] = vidx[i];
        }
    }
}

// ---------------------------------------------------------------------------
// Kernel 2: per-batch softmax over HW of the top-1 similarity values.
// ---------------------------------------------------------------------------
__global__ __launch_bounds__(256) void softmax_kernel(
    const float* __restrict__ wv, float* __restrict__ wsoft)
{
    __shared__ float red[256];
    const int b = blockIdx.x;
    const float* v = wv + (size_t)b * HW;

    float m = -3.402823466e38f;
    for (int n = threadIdx.x; n < HW; n += 256) m = fmaxf(m, v[n]);
    red[threadIdx.x] = m;
    __syncthreads();
    for (int s = 128; s > 0; s >>= 1) {
        if (threadIdx.x < s) red[threadIdx.x] = fmaxf(red[threadIdx.x], red[threadIdx.x + s]);
        __syncthreads();
    }
    m = red[0];
    __syncthreads();

    float sum = 0.0f;
    for (int n = threadIdx.x; n < HW; n += 256) sum += __expf(v[n] - m);
    red[threadIdx.x] = sum;
    __syncthreads();
    for (int s = 128; s > 0; s >>= 1) {
        if (threadIdx.x < s) red[threadIdx.x] += red[threadIdx.x + s];
        __syncthreads();
    }
    const float inv = 1.0f / red[0];

    for (int n = threadIdx.x; n < HW; n += 256)
        wsoft[(size_t)b * HW + n] = __expf(v[n] - m) * inv;
}

// ---------------------------------------------------------------------------
// Kernel 3: gather + conv1x1 ([64x128] W) + masked blend. One 64-thread block
// per (b, n); shared x columns broadcast via LDS. mask==0 positions pass x.
// ---------------------------------------------------------------------------
__global__ __launch_bounds__(64) void conv_blend_kernel(
    const float* __restrict__ x, const float* __restrict__ mask,
    const float* __restrict__ Wm, const float* __restrict__ bias,
    const int* __restrict__ idx, const float* __restrict__ wsoft,
    float* __restrict__ out)
{
    __shared__ float xj[Cc];     // matched support column s[:, j] source
    __shared__ float xn[Cc];     // query column x[:, n]
    __shared__ float wjS;
    __shared__ int   jS;

    const int b = blockIdx.x / HW;
    const int n = blockIdx.x % HW;
    const int o = threadIdx.x;

    const float* xb = x + (size_t)b * Cc * HW;
    const float mn  = mask[(size_t)b * HW + n];

    if (mn == 0.0f) {            // uniform across block: blend passes x through
        out[(size_t)b * Cc * HW + (size_t)o * HW + n] = xb[(size_t)o * HW + n];
        return;
    }

    if (o == 0) {
        const int j = idx[(size_t)b * HW + n];
        jS  = j;
        wjS = wsoft[(size_t)b * HW + n] * (1.0f - mask[(size_t)b * HW + j]);
    }
    __syncthreads();
    const int   j  = jS;
    const float wj = wjS;

    xj[o] = xb[(size_t)o * HW + j];
    xn[o] = xb[(size_t)o * HW + n];      // mask[n]==1 -> q column == x column
    __syncthreads();

    const float* wr = Wm + (size_t)o * (2 * Cc);
    float a1 = 0.0f, a2 = 0.0f;
    #pragma unroll 8
    for (int c = 0; c < Cc; ++c) {
        a1 += wr[c]      * xj[c];        // fuse = w[n] * sel
        a2 += wr[Cc + c] * xn[c];        // q part
    }
    out[(size_t)b * Cc * HW + (size_t)o * HW + n] = bias[o] + wj * a1 + a2;
}

// ---------------------------------------------------------------------------
extern "C" void kernel_launch(void* const* d_in, const int* in_sizes, int n_in,
                              void* d_out, int out_size, void* d_ws, size_t ws_size,
                              hipStream_t stream)
{
    const float* x    = (const float*)d_in[0];
    const float* mask = (const float*)d_in[1];
    const float* Wm   = (const float*)d_in[2];
    const float* bias = (const float*)d_in[3];
    float* out = (float*)d_out;

    char*  ws    = (char*)d_ws;
    float* wv    = (float*)ws;                                    // [B,HW]
    int*   idx   = (int*)  (ws + 1 * Bn * HW * sizeof(float));    // [B,HW]
    float* wsoft = (float*)(ws + 2 * Bn * HW * sizeof(float));    // [B,HW]

    dim3 g1(HW / (16 * 8), Bn);   // 8 query tiles (waves) per block
    simi_argmax_kernel<<<g1, 256, 0, stream>>>(x, mask, wv, idx);
    softmax_kernel<<<Bn, 256, 0, stream>>>(wv, wsoft);
    conv_blend_kernel<<<Bn * HW, 64, 0, stream>>>(x, mask, Wm, bias, idx, wsoft, out);
}